// Pool_32212254720661
// MI455X (gfx1250) — compile-verified
//
#include <hip/hip_runtime.h>
#include <hip/hip_bf16.h>

#define N_NODES 100000
#define N_EDGES 3200000
#define IN_DIM  256
#define K_KEEP  50000       // max(2, int(0.5 * N_NODES))
#define SORT_N  131072      // next pow2 >= N_NODES
#define LOCAL_N 4096        // LDS sort block: 4096 * 8B = 32 KB of LDS

typedef float v2f __attribute__((ext_vector_type(2)));
typedef float v8f __attribute__((ext_vector_type(8)));
typedef unsigned long long u64;
typedef unsigned int u32;

// ---------------------------------------------------------------------------
// K1: scores[i] = sigmoid(dot(h[i,:], W) + b) via V_WMMA_F32_16X16X4_F32.
// One wave handles 16 nodes. A = 16x4 tile of h (lanes 0-15 carry K0/K1,
// lanes 16-31 carry K2/K3 per the CDNA5 f32 A layout). B[k,n] = W[k] for all
// n (broadcast), so every column of D holds the same dot products.
// D layout: lane 0 holds rows 0..7 (VGPR 0..7), lane 16 holds rows 8..15.
// ---------------------------------------------------------------------------
__global__ void scores_wmma_kernel(const float* __restrict__ h,
                                   const float* __restrict__ W,
                                   const float* __restrict__ bias,
                                   float* __restrict__ scores) {
  int wave = blockIdx.x * (blockDim.x >> 5) + (threadIdx.x >> 5);
  int lane = threadIdx.x & 31;
  const int ntiles = N_NODES / 16;  // 6250, exact
  if (wave >= ntiles) return;       // wave-uniform exit: EXEC stays all-1s

  int base = wave * 16;
  int m    = lane & 15;             // node row within tile
  int koff = (lane >> 4) << 1;      // 0 for lanes 0-15, 2 for lanes 16-31

  const float* hrow = h + (size_t)(base + m) * IN_DIM;

  v8f acc = {0.f, 0.f, 0.f, 0.f, 0.f, 0.f, 0.f, 0.f};
  for (int k0 = 0; k0 < IN_DIM; k0 += 4) {
    v2f a, bb;
    a.x  = hrow[k0 + koff];
    a.y  = hrow[k0 + koff + 1];
    bb.x = W[k0 + koff];
    bb.y = W[k0 + koff + 1];
    acc = __builtin_amdgcn_wmma_f32_16x16x4_f32(false, a, false, bb,
                                                (short)0, acc, false, false);
  }

  float bv = bias[0];
  if (lane == 0) {
    #pragma unroll
    for (int r = 0; r < 8; ++r) {
      float x = acc[r] + bv;
      scores[base + r] = 1.0f / (1.0f + __expf(-x));
    }
  } else if (lane == 16) {
    #pragma unroll
    for (int r = 0; r < 8; ++r) {
      float x = acc[r] + bv;
      scores[base + 8 + r] = 1.0f / (1.0f + __expf(-x));
    }
  }
}

// monotone map: float compare == unsigned compare of fkey()
__device__ __forceinline__ u32 fkey(float f) {
  u32 u = __float_as_uint(f);
  return (u & 0x80000000u) ? ~u : (u | 0x80000000u);
}

// K2: pack sort keys. Descending score (invert monotone key) with index in
// low bits for jax top_k tie-breaking (lower index first). Padding sorts last.
__global__ void build_keys_kernel(const float* __restrict__ scores,
                                  u64* __restrict__ keys) {
  int i = blockIdx.x * blockDim.x + threadIdx.x;
  if (i >= SORT_N) return;
  if (i < N_NODES) {
    u32 k = ~fkey(scores[i]);
    keys[i] = ((u64)k << 32) | (u32)i;
  } else {
    keys[i] = ~0ull;
  }
}

// ---------------------------------------------------------------------------
// Bitonic sort, LDS-tiled. Each 4096-element block (32 KB keys) lives in LDS.
// Pair index p -> element i = ((p & ~(j-1)) << 1) | (p & (j-1)), partner i|j.
// Direction uses the GLOBAL element index, so LDS phases compose exactly with
// the global stages into one ascending sort of SORT_N keys.
// ---------------------------------------------------------------------------

// K3a: full local sort: all stages k = 2 .. LOCAL_N (replaces 78 global stages)
__global__ void bitonic_local_sort_kernel(u64* __restrict__ keys) {
  __shared__ u64 s[LOCAL_N];
  u32 base = blockIdx.x * LOCAL_N;
  for (u32 t = threadIdx.x; t < LOCAL_N; t += blockDim.x)
    s[t] = keys[base + t];
  __syncthreads();
  for (u32 k = 2; k <= LOCAL_N; k <<= 1) {
    for (u32 j = k >> 1; j >= 1; j >>= 1) {
      for (u32 p = threadIdx.x; p < LOCAL_N / 2; p += blockDim.x) {
        u32 i = ((p & ~(j - 1)) << 1) | (p & (j - 1));
        u32 l = i | j;
        u64 a = s[i];
        u64 b = s[l];
        bool up = (((base + i) & k) == 0);
        if (up ? (a > b) : (a < b)) { s[i] = b; s[l] = a; }
      }
      __syncthreads();
    }
  }
  for (u32 t = threadIdx.x; t < LOCAL_N; t += blockDim.x)
    keys[base + t] = s[t];
}

// K3b: one global compare-exchange stage (only needed while j >= LOCAL_N)
__global__ void bitonic_stage_kernel(u64* __restrict__ keys, u32 k, u32 j) {
  u32 i = blockIdx.x * blockDim.x + threadIdx.x;
  u32 l = i ^ j;
  if (l > i) {
    u64 a = keys[i];
    u64 b = keys[l];
    bool up = ((i & k) == 0);
    if (up ? (a > b) : (a < b)) { keys[i] = b; keys[l] = a; }
  }
}

// K3c: fused local merge tail for fixed k: stages j = LOCAL_N/2 .. 1 in LDS
__global__ void bitonic_local_merge_kernel(u64* __restrict__ keys, u32 k) {
  __shared__ u64 s[LOCAL_N];
  u32 base = blockIdx.x * LOCAL_N;
  for (u32 t = threadIdx.x; t < LOCAL_N; t += blockDim.x)
    s[t] = keys[base + t];
  __syncthreads();
  for (u32 j = LOCAL_N >> 1; j >= 1; j >>= 1) {
    for (u32 p = threadIdx.x; p < LOCAL_N / 2; p += blockDim.x) {
      u32 i = ((p & ~(j - 1)) << 1) | (p & (j - 1));
      u32 l = i | j;
      u64 a = s[i];
      u64 b = s[l];
      bool up = (((base + i) & k) == 0);
      if (up ? (a > b) : (a < b)) { s[i] = b; s[l] = a; }
    }
    __syncthreads();
  }
  for (u32 t = threadIdx.x; t < LOCAL_N; t += blockDim.x)
    keys[base + t] = s[t];
}

__global__ void zero_member_kernel(unsigned char* __restrict__ member) {
  int i = blockIdx.x * blockDim.x + threadIdx.x;
  if (i < N_NODES) member[i] = 0;
}

// K4: mark selected nodes + emit idx output (as float, exact for idx < 2^24).
__global__ void commit_topk_kernel(const u64* __restrict__ keys,
                                   unsigned char* __restrict__ member,
                                   float* __restrict__ out_idx) {
  int i = blockIdx.x * blockDim.x + threadIdx.x;
  if (i >= K_KEEP) return;
  u32 idx = (u32)(keys[i] & 0xFFFFFFFFu);
  member[idx] = 1;
  out_idx[i] = (float)idx;
}

// K5: new_h[r,:] = h[idx[r],:] * scores[idx[r]]  -- one float4 per thread
__global__ void gather_h_kernel(const u64* __restrict__ keys,
                                const float* __restrict__ scores,
                                const float* __restrict__ h,
                                float* __restrict__ out_h) {
  u32 gid = blockIdx.x * blockDim.x + threadIdx.x;
  const u32 total = K_KEEP * (IN_DIM / 4);  // 3,200,000 float4s
  if (gid >= total) return;
  u32 r = gid >> 6;           // / (IN_DIM/4)
  u32 c = (gid & 63) << 2;    // float offset, 16B aligned
  u32 idx = (u32)(keys[r] & 0xFFFFFFFFu);
  float v = scores[idx];
  float4 x = *(const float4*)(h + (size_t)idx * IN_DIM + c);
  x.x *= v; x.y *= v; x.z *= v; x.w *= v;
  *(float4*)(out_h + (size_t)r * IN_DIM + c) = x;
}

// K6: edge filter, 4 edges per thread; dropped edges -> -1 in both rows.
__global__ void edge_filter_kernel(const int* __restrict__ eidx,
                                   const unsigned char* __restrict__ member,
                                   float* __restrict__ out_e) {
  u32 q = blockIdx.x * blockDim.x + threadIdx.x;
  const u32 total = N_EDGES / 4;  // 800,000
  if (q >= total) return;
  int4 s = ((const int4*)eidx)[q];
  int4 t = ((const int4*)(eidx + N_EDGES))[q];
  float4 os, ot;
  bool k0 = member[s.x] && member[t.x];
  bool k1 = member[s.y] && member[t.y];
  bool k2 = member[s.z] && member[t.z];
  bool k3 = member[s.w] && member[t.w];
  os.x = k0 ? (float)s.x : -1.0f;  ot.x = k0 ? (float)t.x : -1.0f;
  os.y = k1 ? (float)s.y : -1.0f;  ot.y = k1 ? (float)t.y : -1.0f;
  os.z = k2 ? (float)s.z : -1.0f;  ot.z = k2 ? (float)t.z : -1.0f;
  os.w = k3 ? (float)s.w : -1.0f;  ot.w = k3 ? (float)t.w : -1.0f;
  ((float4*)out_e)[q] = os;
  ((float4*)(out_e + N_EDGES))[q] = ot;
}

extern "C" void kernel_launch(void* const* d_in, const int* in_sizes, int n_in,
                              void* d_out, int out_size, void* d_ws, size_t ws_size,
                              hipStream_t stream) {
  (void)in_sizes; (void)n_in; (void)out_size; (void)ws_size;

  // inputs: g(0, unused), h(1), edge_index(2), W(3), b(4)
  const float* h    = (const float*)d_in[1];
  const int*   eidx = (const int*)  d_in[2];
  const float* W    = (const float*)d_in[3];
  const float* bias = (const float*)d_in[4];

  // outputs concatenated flat (float): new_edge_index[2*E], new_h[K*D], idx[K]
  float* out      = (float*)d_out;
  float* out_edge = out;
  float* out_h    = out + (size_t)2 * N_EDGES;
  float* out_idx  = out_h + (size_t)K_KEEP * IN_DIM;

  // workspace layout
  char* ws = (char*)d_ws;
  u64*           keys   = (u64*)ws;                                   // 1 MiB
  float*         scores = (float*)(ws + (size_t)SORT_N * 8);          // 400 KB
  unsigned char* member = (unsigned char*)(ws + (size_t)SORT_N * 8 +
                                           (size_t)N_NODES * 4);      // 100 KB

  // 1) projection scores via WMMA (6250 tiles of 16 nodes, 8 waves/block)
  scores_wmma_kernel<<<(N_NODES / 16 + 7) / 8, 256, 0, stream>>>(h, W, bias, scores);

  // 2) bitonic sort: LDS block sort, then global stages + fused LDS merge tails
  build_keys_kernel<<<SORT_N / 256, 256, 0, stream>>>(scores, keys);
  bitonic_local_sort_kernel<<<SORT_N / LOCAL_N, 256, 0, stream>>>(keys);
  for (u32 k = LOCAL_N * 2; k <= SORT_N; k <<= 1) {
    for (u32 j = k >> 1; j >= LOCAL_N; j >>= 1) {
      bitonic_stage_kernel<<<SORT_N / 256, 256, 0, stream>>>(keys, k, j);
    }
    bitonic_local_merge_kernel<<<SORT_N / LOCAL_N, 256, 0, stream>>>(keys, k);
  }

  // 3) membership + outputs
  zero_member_kernel<<<(N_NODES + 255) / 256, 256, 0, stream>>>(member);
  commit_topk_kernel<<<(K_KEEP + 255) / 256, 256, 0, stream>>>(keys, member, out_idx);
  gather_h_kernel<<<(K_KEEP * (IN_DIM / 4) + 255) / 256, 256, 0, stream>>>(keys, scores, h, out_h);
  edge_filter_kernel<<<(N_EDGES / 4 + 255) / 256, 256, 0, stream>>>(eidx, member, out_edge);
}